// CRDM_24292335026247
// MI455X (gfx1250) — compile-verified
//
#include <hip/hip_runtime.h>

typedef float v2f __attribute__((ext_vector_type(2)));
typedef float v8f __attribute__((ext_vector_type(8)));

#define DIMN  256
#define NAQ   2048
#define NBK   4096
#define NBATCH 8
#define SCALE 0.0625f   // 1/sqrt(256)

// ---------------------------------------------------------------------------
// f32 WMMA: D(16x16) = A(16x4) * B(4x16) + C    (full fp32 precision)
// Lane layout (ISA 7.12.2): l16 = lane&15, hi = lane>>4
//   A frag: a[v] = A[l16][hi*2+v]      (row-major 16xK tile)
//   B frag: b[v] = B[hi*2+v][l16]  ==  Kt[l16][hi*2+v] for B = Kt^T
//   D frag: d[r] = D[r + hi*8][l16]
// ---------------------------------------------------------------------------
__device__ __forceinline__ v8f wmma_k4(v2f a, v2f b, v8f c) {
  return __builtin_amdgcn_wmma_f32_16x16x4_f32(false, a, false, b,
                                               (short)0, c, false, false);
}

// Register-blocked 16x64: one shared A fragment drives 4 B fragments
// (4 consecutive 16-row tiles of the "column" matrix, spaced 16*DIMN).
// 5 loads per 4 WMMAs instead of 2 loads per WMMA.
__device__ __forceinline__ void tile_gemm256_x4(const float* __restrict__ qr,
                                                const float* __restrict__ kr,
                                                v8f acc[4]) {
#pragma unroll 4
  for (int k = 0; k < DIMN; k += 4) {
    v2f a = *(const v2f*)(qr + k);
    acc[0] = wmma_k4(a, *(const v2f*)(kr + k),                 acc[0]);
    acc[1] = wmma_k4(a, *(const v2f*)(kr + 16 * DIMN + k),     acc[1]);
    acc[2] = wmma_k4(a, *(const v2f*)(kr + 32 * DIMN + k),     acc[2]);
    acc[3] = wmma_k4(a, *(const v2f*)(kr + 48 * DIMN + k),     acc[3]);
  }
}

// ---------------------------------------------------------------------------
// Projection: Y[rows x 256] = X @ W^T + bias.  One wave per 16x64 block.
// ---------------------------------------------------------------------------
__global__ __launch_bounds__(256) void proj_kernel(
    const float* __restrict__ X, const float* __restrict__ W,
    const float* __restrict__ bias, float* __restrict__ Y, int rows)
{
  int wave = blockIdx.x * 8 + (threadIdx.x >> 5);
  int mt = wave >> 2;            // row tile
  int ng = wave & 3;             // 4 groups of 4 col tiles (256 = 4*64)
  if (mt * 16 >= rows) return;
  int lane = threadIdx.x & 31;
  int l16 = lane & 15, hi = lane >> 4;
  const float* xr = X + (size_t)(mt * 16 + l16) * DIMN + hi * 2;
  const float* wr = W + (size_t)(ng * 64 + l16) * DIMN + hi * 2;
  v8f acc[4] = {v8f{}, v8f{}, v8f{}, v8f{}};
  tile_gemm256_x4(xr, wr, acc);
#pragma unroll
  for (int t = 0; t < 4; ++t) {
    int n = ng * 64 + t * 16 + l16;
    float bn = bias[n];
#pragma unroll
    for (int r = 0; r < 8; ++r)
      Y[(size_t)(mt * 16 + r + hi * 8) * DIMN + n] = acc[t][r] + bn;
  }
}

// ---------------------------------------------------------------------------
// Pass 1: per-row online (max, sumexp) of scale * Q K^T.
// One wave per 16 query rows; 64 key columns per iteration.
// ---------------------------------------------------------------------------
__global__ __launch_bounds__(128) void attn_pass1(
    const float* __restrict__ Q, const float* __restrict__ K, int NQ, int NK,
    float* __restrict__ rowM, float* __restrict__ rowS)
{
  int wave = blockIdx.x * 4 + (threadIdx.x >> 5);
  int ntq = NQ >> 4;
  int b = wave / ntq, qt = wave % ntq;
  const float* Qb = Q + (size_t)b * NQ * DIMN;
  const float* Kb = K + (size_t)b * NK * DIMN;
  int lane = threadIdx.x & 31, l16 = lane & 15, hi = lane >> 4;
  const float* qr = Qb + (size_t)(qt * 16 + l16) * DIMN + hi * 2;
  float rm[8], rs[8];
#pragma unroll
  for (int r = 0; r < 8; ++r) { rm[r] = -3.4e38f; rs[r] = 0.f; }
  for (int cg = 0; cg < (NK >> 6); ++cg) {
    const float* kr = Kb + (size_t)(cg * 64 + l16) * DIMN + hi * 2;
    v8f acc[4] = {v8f{}, v8f{}, v8f{}, v8f{}};
    tile_gemm256_x4(qr, kr, acc);
#pragma unroll
    for (int t = 0; t < 4; ++t) {
#pragma unroll
      for (int r = 0; r < 8; ++r) {
        float d = acc[t][r] * SCALE;
        float tm = d;                               // tile row max (16 cols)
        tm = fmaxf(tm, __shfl_xor(tm, 8, 16));
        tm = fmaxf(tm, __shfl_xor(tm, 4, 16));
        tm = fmaxf(tm, __shfl_xor(tm, 2, 16));
        tm = fmaxf(tm, __shfl_xor(tm, 1, 16));
        float nm = fmaxf(rm[r], tm);
        float p = __expf(d - nm);
        p += __shfl_xor(p, 8, 16);
        p += __shfl_xor(p, 4, 16);
        p += __shfl_xor(p, 2, 16);
        p += __shfl_xor(p, 1, 16);
        rs[r] = rs[r] * __expf(rm[r] - nm) + p;
        rm[r] = nm;
      }
    }
  }
  if (l16 == 0) {
#pragma unroll
    for (int r = 0; r < 8; ++r) {
      int row = b * NQ + qt * 16 + r + hi * 8;
      rowM[row] = rm[r]; rowS[row] = rs[r];
    }
  }
}

// ---------------------------------------------------------------------------
// Pass 2a: attention_sums = row sums of softmax (recompute logits, divide).
// ---------------------------------------------------------------------------
__global__ __launch_bounds__(128) void attn_rowsum(
    const float* __restrict__ Q, const float* __restrict__ K,
    const float* __restrict__ rowM, const float* __restrict__ rowS,
    int NQ, int NK, float* __restrict__ osum)
{
  int wave = blockIdx.x * 4 + (threadIdx.x >> 5);
  int ntq = NQ >> 4;
  int b = wave / ntq, qt = wave % ntq;
  const float* Qb = Q + (size_t)b * NQ * DIMN;
  const float* Kb = K + (size_t)b * NK * DIMN;
  int lane = threadIdx.x & 31, l16 = lane & 15, hi = lane >> 4;
  const float* qr = Qb + (size_t)(qt * 16 + l16) * DIMN + hi * 2;
  float rm[8], sv[8], accs[8];
#pragma unroll
  for (int r = 0; r < 8; ++r) {
    int row = b * NQ + qt * 16 + r + hi * 8;
    rm[r] = rowM[row]; sv[r] = rowS[row]; accs[r] = 0.f;
  }
  for (int cg = 0; cg < (NK >> 6); ++cg) {
    const float* kr = Kb + (size_t)(cg * 64 + l16) * DIMN + hi * 2;
    v8f acc[4] = {v8f{}, v8f{}, v8f{}, v8f{}};
    tile_gemm256_x4(qr, kr, acc);
#pragma unroll
    for (int t = 0; t < 4; ++t)
#pragma unroll
      for (int r = 0; r < 8; ++r)
        accs[r] += __expf(acc[t][r] * SCALE - rm[r]) / sv[r];
  }
#pragma unroll
  for (int r = 0; r < 8; ++r) {
    float s = accs[r];
    s += __shfl_xor(s, 8, 16);
    s += __shfl_xor(s, 4, 16);
    s += __shfl_xor(s, 2, 16);
    s += __shfl_xor(s, 1, 16);
    if (l16 == 0) osum[b * NQ + qt * 16 + r + hi * 8] = s;
  }
}

// ---------------------------------------------------------------------------
// Pass 2b: column sums of softmax(Q K^T) over all NQ rows.
// One wave per 64 columns (4 tiles, fixed); iterates all query row tiles.
// ---------------------------------------------------------------------------
__global__ __launch_bounds__(128) void attn_colsum(
    const float* __restrict__ Q, const float* __restrict__ K,
    const float* __restrict__ rowM, const float* __restrict__ rowS,
    int NQ, int NK, float* __restrict__ ocol)
{
  int wave = blockIdx.x * 4 + (threadIdx.x >> 5);
  int ncg = NK >> 6;
  int b = wave / ncg, cg = wave % ncg;
  const float* Qb = Q + (size_t)b * NQ * DIMN;
  const float* Kb = K + (size_t)b * NK * DIMN;
  int lane = threadIdx.x & 31, l16 = lane & 15, hi = lane >> 4;
  const float* kr = Kb + (size_t)(cg * 64 + l16) * DIMN + hi * 2;
  float csum[4] = {0.f, 0.f, 0.f, 0.f};
  for (int qt = 0; qt < (NQ >> 4); ++qt) {
    const float* qr = Qb + (size_t)(qt * 16 + l16) * DIMN + hi * 2;
    float rm[8], sv[8];
#pragma unroll
    for (int r = 0; r < 8; ++r) {
      int row = b * NQ + qt * 16 + r + hi * 8;
      rm[r] = rowM[row]; sv[r] = rowS[row];
    }
    v8f acc[4] = {v8f{}, v8f{}, v8f{}, v8f{}};
    tile_gemm256_x4(qr, kr, acc);
#pragma unroll
    for (int t = 0; t < 4; ++t)
#pragma unroll
      for (int r = 0; r < 8; ++r)
        csum[t] += __expf(acc[t][r] * SCALE - rm[r]) / sv[r];
  }
#pragma unroll
  for (int t = 0; t < 4; ++t) {
    float c = csum[t];
    c += __shfl_xor(c, 16, 32);       // combine the two half-wave row groups
    if (hi == 0) ocol[b * NK + cg * 64 + t * 16 + l16] = c;
  }
}

// ---------------------------------------------------------------------------
// Bitonic argsort: 2048 keys per batch, descending, ties -> ascending index
// (matches stable argsort(-x)).  One 1024-thread block per batch, 16 KB LDS.
// ---------------------------------------------------------------------------
__global__ __launch_bounds__(1024) void sort_kernel(
    const float* __restrict__ keys, int* __restrict__ oidx)
{
  __shared__ float sk[NAQ];
  __shared__ int   si[NAQ];
  int b = blockIdx.x;
  for (int i = threadIdx.x; i < NAQ; i += blockDim.x) {
    sk[i] = keys[b * NAQ + i]; si[i] = i;
  }
  __syncthreads();
  for (int k = 2; k <= NAQ; k <<= 1) {
    for (int j = k >> 1; j > 0; j >>= 1) {
      for (int t = threadIdx.x; t < NAQ; t += blockDim.x) {
        int p = t ^ j;
        if (p > t) {
          float ka = sk[t], kb2 = sk[p];
          int   ia = si[t], ib  = si[p];
          // "a precedes b" in descending order with index tie-break
          bool before = (ka > kb2) || (ka == kb2 && ia < ib);
          bool up = ((t & k) == 0);
          if (up ? !before : before) {
            sk[t] = kb2; sk[p] = ka; si[t] = ib; si[p] = ia;
          }
        }
      }
      __syncthreads();
    }
  }
  for (int i = threadIdx.x; i < NAQ; i += blockDim.x)
    oidx[b * NAQ + i] = si[i];
}

// ---------------------------------------------------------------------------
// Gather: selected_B[b, i, :] = B[b, idx[b, i], :]
// ---------------------------------------------------------------------------
__global__ __launch_bounds__(64) void gather_kernel(
    const float* __restrict__ B, const int* __restrict__ idx,
    float* __restrict__ out)
{
  int row = blockIdx.x;                 // 0 .. 16383
  int b = row >> 11;                    // / NAQ
  int src = idx[row];
  const float4* s = (const float4*)(B + ((size_t)b * NBK + src) * DIMN);
  float4* d = (float4*)(out + (size_t)row * DIMN);
  d[threadIdx.x] = s[threadIdx.x];
}

// ---------------------------------------------------------------------------
extern "C" void kernel_launch(void* const* d_in, const int* in_sizes, int n_in,
                              void* d_out, int out_size, void* d_ws, size_t ws_size,
                              hipStream_t stream) {
  (void)in_sizes; (void)n_in; (void)out_size; (void)ws_size;
  const float* A   = (const float*)d_in[0];
  const float* B   = (const float*)d_in[1];
  const float* WqA = (const float*)d_in[2];  const float* bqA = (const float*)d_in[3];
  const float* WkA = (const float*)d_in[4];  const float* bkA = (const float*)d_in[5];
  const float* WvA = (const float*)d_in[6];  const float* bvA = (const float*)d_in[7];
  const float* WqB = (const float*)d_in[8];  const float* bqB = (const float*)d_in[9];
  const float* WkB = (const float*)d_in[10]; const float* bkB = (const float*)d_in[11];
  // WvB/bvB (d_in[12],[13]) unused in reference; threshold (d_in[14]) always passes.

  float* out = (float*)d_out;
  float* ws  = (float*)d_ws;

  const int rowsA = NBATCH * NAQ;   // 16384
  const int rowsB = NBATCH * NBK;   // 32768

  // workspace layout (floats)
  float* qa = ws;                                // Q_A  [rowsA x 256]
  float* kb = qa + (size_t)rowsA * DIMN;         // K_B  [rowsB x 256]
  float* qb = kb + (size_t)rowsB * DIMN;         // Q_B  [rowsB x 256]
  float* ka = qb + (size_t)rowsB * DIMN;         // K_A  [rowsA x 256]
  float* mA = ka + (size_t)rowsA * DIMN;         // A2B row max  [rowsA]
  float* sA = mA + rowsA;                        // A2B row sumexp
  float* mB = sA + rowsA;                        // B2A row max  [rowsB]
  float* sB = mB + rowsB;                        // B2A row sumexp
  float* colsum = sB + rowsB;                    // B2A col sums [rowsA]
  int*   idx    = (int*)(colsum + rowsA);        // sorted indices [rowsA]

  // output layout: selected_A | selected_B | (empty) | (empty) | attention_sums
  float* selA  = out;
  float* selB  = out + (size_t)rowsA * DIMN;
  float* asums = selB + (size_t)rowsA * DIMN;

  // 1) projections (V_A written straight to selected_A); 1 wave per 16x64 block
  proj_kernel<<<rowsA / 32, 256, 0, stream>>>(A, WqA, bqA, qa, rowsA);
  proj_kernel<<<rowsB / 32, 256, 0, stream>>>(B, WkB, bkB, kb, rowsB);
  proj_kernel<<<rowsB / 32, 256, 0, stream>>>(B, WqB, bqB, qb, rowsB);
  proj_kernel<<<rowsA / 32, 256, 0, stream>>>(A, WkA, bkA, ka, rowsA);
  proj_kernel<<<rowsA / 32, 256, 0, stream>>>(A, WvA, bvA, selA, rowsA);

  // 2) softmax row stats
  attn_pass1<<<(NBATCH * (NAQ / 16)) / 4, 128, 0, stream>>>(qa, kb, NAQ, NBK, mA, sA);
  attn_pass1<<<(NBATCH * (NBK / 16)) / 4, 128, 0, stream>>>(qb, ka, NBK, NAQ, mB, sB);

  // 3) attention_sums (row sums of softmax A2B)
  attn_rowsum<<<(NBATCH * (NAQ / 16)) / 4, 128, 0, stream>>>(qa, kb, mA, sA, NAQ, NBK, asums);

  // 4) column sums of softmax B2A (argsort keys); 1 wave per 64 columns
  attn_colsum<<<(NBATCH * (NAQ / 64)) / 4, 128, 0, stream>>>(qb, ka, mB, sB, NBK, NAQ, colsum);

  // 5) per-batch descending argsort of 2048 keys
  sort_kernel<<<NBATCH, 1024, 0, stream>>>(colsum, idx);

  // 6) gather selected_B rows
  gather_kernel<<<rowsA, 64, 0, stream>>>(B, idx, selB);
}